// SAGEConv_12481174963003
// MI455X (gfx1250) — compile-verified
//
#include <hip/hip_runtime.h>
#include <hip/hip_bf16.h>

typedef __attribute__((ext_vector_type(2))) float v2f;
typedef __attribute__((ext_vector_type(8))) float v8f;

#define D_FEAT 64

// ---------------------------------------------------------------------------
// Kernel 1: zero the scratch accumulator (agg[N*64] followed by degree[N]).
// ---------------------------------------------------------------------------
__global__ void sage_zero_kernel(float* __restrict__ p, size_t n) {
    size_t i = (size_t)blockIdx.x * blockDim.x + threadIdx.x;
    if (i < n) p[i] = 0.0f;
}

// ---------------------------------------------------------------------------
// Kernel 2: edge scatter. 64 lanes cooperate on one edge (one lane per
// feature), so the gather of x[src] and the atomic adds into agg[dst] are
// fully coalesced 256B transactions. agg (25.6 MB) fits in the 192 MB L2,
// so the atomic RMW traffic stays on-chip.
// ---------------------------------------------------------------------------
__global__ void sage_scatter_kernel(const float* __restrict__ x,
                                    const int* __restrict__ src,
                                    const int* __restrict__ dst,
                                    float* __restrict__ agg,
                                    float* __restrict__ deg,
                                    int n_edges) {
    int e = blockIdx.x * (blockDim.x >> 6) + (threadIdx.x >> 6);
    int f = threadIdx.x & 63;
    if (e >= n_edges) return;
    int s = src[e];
    int d = dst[e];
    atomicAdd(&agg[(size_t)d * D_FEAT + f], x[(size_t)s * D_FEAT + f]);
    if (f == 0) atomicAdd(&deg[d], 1.0f);
}

// ---------------------------------------------------------------------------
// Kernel 3: fused normalize + dual GEMM + bias via V_WMMA_F32_16X16X4_F32.
//
// One wave (32 lanes) owns a 16-node tile. A-fragments (16x4 f32 per the
// CDNA5 layout: v0 = K=k0+2*hi, v1 = K=k0+1+2*hi, lanes 0-15 hold M=0..15)
// for both x and agg/deg are preloaded as per-lane float2s and reused across
// all 4 output column tiles. B-fragments come straight from the torch-style
// [out,in] weight: B[k][col] = w[col*64+k], which is also a contiguous
// per-lane float2. Accumulator init = bias (depends only on N = lane&15).
// Per wave: 4 col tiles x 2 matrices x 16 K-steps = 128 WMMAs.
// Tile guard is wave-uniform so EXEC is all-ones inside the WMMA region.
// ---------------------------------------------------------------------------
__global__ void sage_wmma_gemm_kernel(const float* __restrict__ x,
                                      const float* __restrict__ agg,
                                      const float* __restrict__ deg,
                                      const float* __restrict__ w_neigh,
                                      const float* __restrict__ w_self,
                                      const float* __restrict__ b_self,
                                      float* __restrict__ out,
                                      int n_tiles) {
    const int lane = threadIdx.x & 31;
    const int wave = threadIdx.x >> 5;
    const int tile = blockIdx.x * (blockDim.x >> 5) + wave;
    if (tile >= n_tiles) return;   // uniform per wave

    const int lo = lane & 15;
    const int hi = lane >> 4;
    const int row = tile * 16 + lo;

    const float* xrow = x   + (size_t)row * D_FEAT + 2 * hi;
    const float* arow = agg + (size_t)row * D_FEAT + 2 * hi;
    const float dgr = deg[row];
    const float inv = 1.0f / (dgr > 1.0f ? dgr : 1.0f);

    // Preload all 16 K-step A-fragments for both matrices (degree norm fused).
    v2f ax[16], an[16];
#pragma unroll
    for (int s = 0; s < 16; ++s) {
        ax[s] = *(const v2f*)(xrow + 4 * s);
        v2f t = *(const v2f*)(arow + 4 * s);
        t.x *= inv;
        t.y *= inv;
        an[s] = t;
    }

#pragma unroll
    for (int j = 0; j < 4; ++j) {      // output column tiles (N = j*16..j*16+15)
        const int col = j * 16 + lo;
        const float bias = b_self[col];
        v8f acc;
#pragma unroll
        for (int r = 0; r < 8; ++r) acc[r] = bias;

        const float* ws = w_self  + (size_t)col * D_FEAT + 2 * hi;
        const float* wn = w_neigh + (size_t)col * D_FEAT + 2 * hi;

#pragma unroll
        for (int s = 0; s < 16; ++s) {
            v2f b = *(const v2f*)(ws + 4 * s);
            acc = __builtin_amdgcn_wmma_f32_16x16x4_f32(
                false, ax[s], false, b, (short)0, acc, false, false);
        }
#pragma unroll
        for (int s = 0; s < 16; ++s) {
            v2f b = *(const v2f*)(wn + 4 * s);
            acc = __builtin_amdgcn_wmma_f32_16x16x4_f32(
                false, an[s], false, b, (short)0, acc, false, false);
        }

        // D layout: VGPR r, lane -> out[M = r + 8*hi][N = lo]
#pragma unroll
        for (int r = 0; r < 8; ++r) {
            out[(size_t)(tile * 16 + r + 8 * hi) * D_FEAT + col] = acc[r];
        }
    }
}

// ---------------------------------------------------------------------------
// Host launcher (graph-capture safe: kernels only, all on `stream`).
// ---------------------------------------------------------------------------
extern "C" void kernel_launch(void* const* d_in, const int* in_sizes, int n_in,
                              void* d_out, int out_size, void* d_ws, size_t ws_size,
                              hipStream_t stream) {
    const float* x       = (const float*)d_in[0];
    const int*   eidx    = (const int*)d_in[1];
    const float* w_neigh = (const float*)d_in[2];
    const float* w_self  = (const float*)d_in[3];
    const float* b_self  = (const float*)d_in[4];
    float*       out     = (float*)d_out;

    const int n_nodes = in_sizes[0] / D_FEAT;   // 100000
    const int n_edges = in_sizes[1] / 2;        // 1250000

    float* agg = (float*)d_ws;                        // [n_nodes * 64]
    float* deg = agg + (size_t)n_nodes * D_FEAT;      // [n_nodes]

    // 1) zero agg + degree
    size_t nz = (size_t)n_nodes * D_FEAT + n_nodes;
    sage_zero_kernel<<<(unsigned)((nz + 255) / 256), 256, 0, stream>>>(agg, nz);

    // 2) scatter-add (4 edges per 256-thread block)
    const int* src = eidx;
    const int* dst = eidx + n_edges;
    sage_scatter_kernel<<<(n_edges + 3) / 4, 256, 0, stream>>>(x, src, dst, agg, deg, n_edges);

    // 3) fused normalize + WMMA GEMM + bias (8 waves/block, 16 nodes/wave)
    const int n_tiles = n_nodes / 16;           // 6250 (exact)
    sage_wmma_gemm_kernel<<<(n_tiles + 7) / 8, 256, 0, stream>>>(
        x, agg, deg, w_neigh, w_self, b_self, out, n_tiles);
}